// HypergraphLearing_89635967467803
// MI455X (gfx1250) — compile-verified
//
#include <hip/hip_runtime.h>

typedef __attribute__((ext_vector_type(2))) float v2f;
typedef __attribute__((ext_vector_type(8))) float v8f;

#define B_   8
#define C_   64
#define N_   512
#define T_   288
#define K_   32
#define TT   16
#define LN_EPS 1e-5f

__device__ __forceinline__ v8f wmma4(v2f a, v2f b, v8f c) {
  // D = A(16x4) * B(4x16) + C, full fp32
  return __builtin_amdgcn_wmma_f32_16x16x4_f32(false, a, false, b,
                                               (short)0, c, false, false);
}

__device__ __forceinline__ v8f vzero8() {
  v8f r;
#pragma unroll
  for (int g = 0; g < 8; ++g) r[g] = 0.0f;
  return r;
}

// One workgroup = one (b, 16-wide t-tile). 256 threads = 8 waves.
// LDS: H/O buffer [K=32][C=64][TT=16] f32 = 128 KB (dynamic).
__global__ void __launch_bounds__(256)
hyper_fused(const float* __restrict__ x, const float* __restrict__ he,
            const float* __restrict__ em, float* __restrict__ out)
{
  extern __shared__ float smem[];              // [32][64][16]
  const int lane = threadIdx.x & 31;
  const int wave = threadIdx.x >> 5;
  const int lh   = lane & 15;                  // 0..15  (M or N index in frags)
  const int hi   = lane >> 4;                  // 0/1    (half-wave -> K group)
  const int b    = blockIdx.x / (T_ / TT);
  const int t0   = (blockIdx.x % (T_ / TT)) * TT;
  const int cb   = wave * 8;                   // this wave's channel block

  // ---------------- Stage 1: H[k,c,t] = sum_n HEs[n,k] * x[b,c,n,t] --------
  v8f acc[8][2];
#pragma unroll
  for (int ci = 0; ci < 8; ++ci)
#pragma unroll
    for (int kt = 0; kt < 2; ++kt)
      acc[ci][kt] = vzero8();

  for (int n = 0; n < N_; n += 4) {
    // A frags: A[m,k] = HEs[n+k, kbase+m]  (lane=m; VGPR0:k=2*hi, VGPR1:k=2*hi+1)
    v2f a0, a1;
    a0.x = he[(n + 2 * hi    ) * K_ +      lh];
    a0.y = he[(n + 2 * hi + 1) * K_ +      lh];
    a1.x = he[(n + 2 * hi    ) * K_ + 16 + lh];
    a1.y = he[(n + 2 * hi + 1) * K_ + 16 + lh];
#pragma unroll
    for (int ci = 0; ci < 8; ++ci) {
      const int c = cb + ci;
      const float* xp = x + (((size_t)(b * C_ + c) * N_ + n) * T_ + t0);
      v2f bf;                                  // B[k,t] = x[b,c,n+k,t0+t]
      bf.x = xp[(2 * hi    ) * T_ + lh];
      bf.y = xp[(2 * hi + 1) * T_ + lh];
      acc[ci][0] = wmma4(a0, bf, acc[ci][0]);
      acc[ci][1] = wmma4(a1, bf, acc[ci][1]);
    }
  }
  // spill H tiles to LDS (D layout: VGPR g -> k = 16*kt + g + 8*hi, col = lh)
#pragma unroll
  for (int ci = 0; ci < 8; ++ci)
#pragma unroll
    for (int kt = 0; kt < 2; ++kt)
#pragma unroll
      for (int g = 0; g < 8; ++g) {
        const int k = kt * 16 + g + 8 * hi;
        smem[(k * C_ + cb + ci) * TT + lh] = acc[ci][kt][g];
      }
  __syncthreads();

  // ---------------- Stage 2: O = relu(E @ H) + H  (in place, per-channel) --
  v2f ea[8][2];                                // edge_map A-frags, hoisted
#pragma unroll
  for (int js = 0; js < 8; ++js)
#pragma unroll
    for (int kt = 0; kt < 2; ++kt) {
      ea[js][kt].x = em[(kt * 16 + lh) * K_ + 4 * js + 2 * hi];
      ea[js][kt].y = em[(kt * 16 + lh) * K_ + 4 * js + 2 * hi + 1];
    }
#pragma unroll
  for (int ci = 0; ci < 8; ++ci) {
    const int c = cb + ci;
    v2f hb[8];
#pragma unroll
    for (int js = 0; js < 8; ++js) {
      hb[js].x = smem[((4 * js + 2 * hi    ) * C_ + c) * TT + lh];
      hb[js].y = smem[((4 * js + 2 * hi + 1) * C_ + c) * TT + lh];
    }
    v8f m0 = vzero8(), m1 = vzero8();
#pragma unroll
    for (int js = 0; js < 8; ++js) {
      m0 = wmma4(ea[js][0], hb[js], m0);
      m1 = wmma4(ea[js][1], hb[js], m1);
    }
#pragma unroll
    for (int kt = 0; kt < 2; ++kt)
#pragma unroll
      for (int g = 0; g < 8; ++g) {
        const int k   = kt * 16 + g + 8 * hi;
        const int idx = (k * C_ + c) * TT + lh;
        const float hv = smem[idx];
        const float mv = (kt == 0) ? m0[g] : m1[g];
        smem[idx] = fmaxf(mv, 0.0f) + hv;     // hyper_out
      }
  }
  __syncthreads();

  // ---------------- Stage 3: z = relu(HEs @ O) + x ; fused LayerNorm over c
  const int nt0 = wave * 4;                    // 4 n-tiles of 16 per wave
  v2f a3[4][8];                                // HEs A-frags, hoisted
#pragma unroll
  for (int q = 0; q < 4; ++q) {
    const int nb = (nt0 + q) * 16;
#pragma unroll
    for (int js = 0; js < 8; ++js) {
      a3[q][js].x = he[(nb + lh) * K_ + 4 * js + 2 * hi];
      a3[q][js].y = he[(nb + lh) * K_ + 4 * js + 2 * hi + 1];
    }
  }

  float S[4][8] = {}, Q[4][8] = {};
  // pass 1: accumulate per-(n,t) mean / mean-square stats across c
  for (int c = 0; c < C_; ++c) {
    v2f ob[8];
#pragma unroll
    for (int js = 0; js < 8; ++js) {
      ob[js].x = smem[((4 * js + 2 * hi    ) * C_ + c) * TT + lh];
      ob[js].y = smem[((4 * js + 2 * hi + 1) * C_ + c) * TT + lh];
    }
#pragma unroll
    for (int q = 0; q < 4; ++q) {
      v8f a = vzero8();
#pragma unroll
      for (int js = 0; js < 8; ++js) a = wmma4(a3[q][js], ob[js], a);
      const float* xp =
          x + (((size_t)(b * C_ + c) * N_ + (nt0 + q) * 16 + 8 * hi) * T_ + t0 + lh);
#pragma unroll
      for (int g = 0; g < 8; ++g) {
        const float z = fmaxf(a[g], 0.0f) + xp[g * T_];
        S[q][g] += z;
        Q[q][g] += z * z;
      }
    }
  }
  float mu[4][8], rs[4][8];
#pragma unroll
  for (int q = 0; q < 4; ++q)
#pragma unroll
    for (int g = 0; g < 8; ++g) {
      mu[q][g] = S[q][g] * (1.0f / C_);
      float var = Q[q][g] * (1.0f / C_) - mu[q][g] * mu[q][g];
      rs[q][g] = rsqrtf(fmaxf(var, 0.0f) + LN_EPS);
    }

  // pass 2: recompute z (O is LDS-hot, x is L0/L2-hot), normalize, store
  for (int c = 0; c < C_; ++c) {
    v2f ob[8];
#pragma unroll
    for (int js = 0; js < 8; ++js) {
      ob[js].x = smem[((4 * js + 2 * hi    ) * C_ + c) * TT + lh];
      ob[js].y = smem[((4 * js + 2 * hi + 1) * C_ + c) * TT + lh];
    }
#pragma unroll
    for (int q = 0; q < 4; ++q) {
      v8f a = vzero8();
#pragma unroll
      for (int js = 0; js < 8; ++js) a = wmma4(a3[q][js], ob[js], a);
      const size_t base =
          ((size_t)(b * C_ + c) * N_ + (nt0 + q) * 16 + 8 * hi) * T_ + t0 + lh;
      const float* xp = x + base;
      float*       op = out + base;
#pragma unroll
      for (int g = 0; g < 8; ++g) {
        const float z = fmaxf(a[g], 0.0f) + xp[g * T_];
        op[g * T_] = (z - mu[q][g]) * rs[q][g];
      }
    }
  }
}

extern "C" void kernel_launch(void* const* d_in, const int* in_sizes, int n_in,
                              void* d_out, int out_size, void* d_ws, size_t ws_size,
                              hipStream_t stream) {
  (void)in_sizes; (void)n_in; (void)out_size; (void)d_ws; (void)ws_size;
  const float* x   = (const float*)d_in[0];   // [8,64,512,288]
  const float* he  = (const float*)d_in[1];   // [512,32]
  const float* em  = (const float*)d_in[2];   // [32,32]
  float*       out = (float*)d_out;           // [8,64,512,288]

  const dim3 grid(B_ * (T_ / TT));            // 144 workgroups
  const dim3 block(256);                      // 8 wave32 waves
  const size_t shmem = (size_t)K_ * C_ * TT * sizeof(float);  // 128 KB
  hyper_fused<<<grid, block, shmem, stream>>>(x, he, em, out);
}